// Net_34857954574879
// MI455X (gfx1250) — compile-verified
//
#include <hip/hip_runtime.h>
#include <hip/hip_bf16.h>

typedef float v2f __attribute__((ext_vector_type(2)));
typedef float v8f __attribute__((ext_vector_type(8)));

#define NN    800
#define EE    25600          // total edges (2*E_HALF)
#define HC    32             // H1 == H2 == 32
#define PLANE 640000         // 800*800
#define NP    4096

// ---------------- utility ----------------
__device__ __forceinline__ float blockReduceSum256(float v) {
    __shared__ float s[256];
    __syncthreads();
    s[threadIdx.x] = v;
    __syncthreads();
    for (int o = 128; o > 0; o >>= 1) {
        if ((int)threadIdx.x < o) s[threadIdx.x] += s[threadIdx.x + o];
        __syncthreads();
    }
    float r = s[0];
    __syncthreads();
    return r;
}

__global__ void zero_f32(float* __restrict__ p, long n) {
    long i = (long)blockIdx.x * blockDim.x + threadIdx.x;
    long stride = (long)gridDim.x * blockDim.x;
    for (; i < n; i += stride) p[i] = 0.0f;
}

__global__ void fill_f32(float* __restrict__ p, int n, float v) {
    int i = blockIdx.x * blockDim.x + threadIdx.x;
    if (i < n) p[i] = v;
}

// ---------------- stage A: GCN ----------------
__global__ void emb_kernel(const int* __restrict__ x, const float* __restrict__ emb,
                           float* __restrict__ h) {
    int t = blockIdx.x * blockDim.x + threadIdx.x;
    if (t >= NN * HC) return;
    h[t] = emb[x[t >> 5] * HC + (t & 31)];
}

__global__ void deg_kernel(const int* __restrict__ ei, float* __restrict__ deg) {
    int e = blockIdx.x * blockDim.x + threadIdx.x;
    if (e >= EE) return;
    atomicAdd(&deg[ei[EE + e]], 1.0f);   // row = ei[1]
}

__global__ void matmul32(const float* __restrict__ h, const float* __restrict__ w,
                         float* __restrict__ o) {
    int t = blockIdx.x * blockDim.x + threadIdx.x;
    if (t >= NN * HC) return;
    int i = t >> 5, dout = t & 31;
    float acc = 0.f;
#pragma unroll
    for (int c = 0; c < HC; ++c) acc += h[i * HC + c] * w[c * HC + dout];
    o[t] = acc;
}

__global__ void agg_init(const float* __restrict__ hw, const float* __restrict__ deg,
                         const float* __restrict__ gb, float* __restrict__ agg) {
    int t = blockIdx.x * blockDim.x + threadIdx.x;
    if (t >= NN * HC) return;
    int i = t >> 5, d = t & 31;
    agg[t] = gb[d] + hw[t] / deg[i];     // self-loop: norm = 1/deg
}

__global__ void scatter_gcn(const int* __restrict__ ei, const float* __restrict__ hw,
                            const float* __restrict__ deg, float* __restrict__ agg) {
    int e = blockIdx.x * blockDim.x + threadIdx.x;
    if (e >= EE) return;
    int col = ei[e];          // ei[0]
    int row = ei[EE + e];     // ei[1]
    float nrm = rsqrtf(deg[row] * deg[col]);
#pragma unroll
    for (int d = 0; d < HC; ++d)
        atomicAdd(&agg[row * HC + d], hw[col * HC + d] * nrm);
}

// graph_norm (unmasked) + relu, per-channel block
__global__ void gn_relu_kernel(const float* __restrict__ agg, const float* __restrict__ w,
                               const float* __restrict__ b, const float* __restrict__ aa,
                               float* __restrict__ hout) {
    int d = blockIdx.x;               // channel
    int tid = threadIdx.x;            // 256 threads
    float vals[4];
    float s = 0.f;
#pragma unroll
    for (int t = 0; t < 4; ++t) {
        int i = tid + t * 256;
        vals[t] = (i < NN) ? agg[i * HC + d] : 0.f;
        s += vals[t];
    }
    float m = blockReduceSum256(s) * (1.0f / NN);
    float am = aa[d] * m;
    float s2 = 0.f;
#pragma unroll
    for (int t = 0; t < 4; ++t) {
        int i = tid + t * 256;
        float xc = (i < NN) ? (vals[t] - am) : 0.f;
        vals[t] = xc;
        s2 += xc * xc;
    }
    float var = blockReduceSum256(s2) * (1.0f / NN);
    float rstd = rsqrtf(var + 1e-5f);
    float wv = w[d], bv = b[d];
#pragma unroll
    for (int t = 0; t < 4; ++t) {
        int i = tid + t * 256;
        if (i < NN) hout[i * HC + d] = fmaxf(wv * vals[t] * rstd + bv, 0.f);
    }
}

// ---------------- stage B: edge MLPs ----------------
__global__ void edge_mlp(const int* __restrict__ ei, const float* __restrict__ h,
                         const float* __restrict__ w1, const float* __restrict__ b1,
                         const float* __restrict__ w2, const float* __restrict__ b2,
                         float* __restrict__ hv, float* __restrict__ mulb) {
    int t = blockIdx.x * blockDim.x + threadIdx.x;
    if (t >= EE * HC) return;
    int e = t >> 5, o = t & 31;
    int i0 = ei[e], i1 = ei[EE + e];
    float a1 = b1[o], a2 = b2[o];
#pragma unroll
    for (int c = 0; c < HC; ++c) {
        float v = h[i0 * HC + c];
        a1 += v * w1[c * HC + o];
        a2 += v * w2[c * HC + o];
    }
#pragma unroll
    for (int c = 0; c < HC; ++c) {
        float v = h[i1 * HC + c];
        a1 += v * w1[(HC + c) * HC + o];
        a2 += v * w2[(HC + c) * HC + o];
    }
    hv[t] = fmaxf(a1, 0.f);
    mulb[t] = fmaxf(a2, 0.f);
}

// ---------------- stage C: build dense channel-major tensors ----------------
// Both GEMM operands are stored K-pair-interleaved so each WMMA fragment is one
// 8-byte-per-lane load (global_load_b64):
//   A(m=i, K=j)  at At[d][j>>1][i][j&1]   (layout [k/2][m][2], row = 1600 floats)
//   B(K=i, n=j)  at Mp[d][i>>1][j][i&1]   (layout [k/2][n][2])
__global__ void scatter_dense(const int* __restrict__ ei, const float* __restrict__ hv,
                              const float* __restrict__ mulb,
                              float* __restrict__ At, float* __restrict__ Mp,
                              float* __restrict__ ind, unsigned int* __restrict__ Abits) {
    int t = blockIdx.x * blockDim.x + threadIdx.x;
    if (t >= EE * HC) return;
    int e = t >> 5, d = t & 31;
    int i = ei[e];           // ei[0]
    int j = ei[EE + e];      // ei[1]
    atomicAdd(&At[(size_t)d * PLANE + (j >> 1) * (2 * NN) + i * 2 + (j & 1)], hv[t]);
    atomicAdd(&Mp[(size_t)d * PLANE + (i >> 1) * (2 * NN) + j * 2 + (i & 1)], mulb[t]);
    if (d == 0) {
        atomicAdd(&ind[i * NN + j], 1.0f);
        atomicOr(&Abits[i * 25 + (j >> 5)], 1u << (j & 31));
    }
}

// U = A | (A@A > 0), using row bitsets (A is symmetric by construction of ei)
__global__ void u_kernel(const unsigned int* __restrict__ Abits, unsigned char* __restrict__ U) {
    int p = blockIdx.x * blockDim.x + threadIdx.x;
    if (p >= PLANE) return;
    int i = p / NN, j = p - i * NN;
    const unsigned int* ri = Abits + i * 25;
    const unsigned int* rj = Abits + j * 25;
    unsigned int o = 0;
#pragma unroll
    for (int w = 0; w < 25; ++w) o |= ri[w] & rj[w];
    unsigned int a = (ri[j >> 5] >> (j & 31)) & 1u;
    U[p] = (a | (o != 0u)) ? 1 : 0;
}

__global__ void cnt_kernel(const unsigned char* __restrict__ U, float* __restrict__ cnt) {
    float s = 0.f;
    for (int p = blockIdx.x * blockDim.x + threadIdx.x; p < PLANE; p += gridDim.x * blockDim.x)
        s += (float)U[p];
    s = blockReduceSum256(s);
    if (threadIdx.x == 0) atomicAdd(cnt, s);
}

// ---------------- stage D: the big batched fp32 GEMM via WMMA ----------------
// C[d] (800x800) = A[d] x B[d], operands in K-pair-interleaved layout.
// grid (25 m-tiles of 32, 5 n-groups of 160, 32 channels); block = 2 waves,
// each wave owns a 32x80 C tile = 2x5 fragments of V_WMMA_F32_16X16X4_F32.
// Per K-step of 4: 7 global_load_b64 feed 10 WMMAs.
__global__ __launch_bounds__(64) void einsum_wmma(const float* __restrict__ Atg,
                                                  const float* __restrict__ Bg,
                                                  float* __restrict__ Cg) {
    int d = blockIdx.z;
    const float* a = Atg + (size_t)d * PLANE;
    const float* b = Bg + (size_t)d * PLANE;
    float* c = Cg + (size_t)d * PLANE;
    int wave = threadIdx.x >> 5;   // 0..1
    int lane = threadIdx.x & 31;
    int half = lane >> 4;          // lanes 16-31 take K+2,K+3 (ISA fragment layout)
    int l16 = lane & 15;
    int m0 = blockIdx.x * 32;
    int n0 = blockIdx.y * 160 + wave * 80;

    // packed row = 1600 floats; lane base points at its (K=k+2*half, K+1) pair
    const float* ak = a + half * (2 * NN) + (m0 + l16) * 2;
    const float* bk = b + half * (2 * NN) + (n0 + l16) * 2;

    v8f acc[2][5] = {};
    for (int k = 0; k < NN; k += 4) {
        v2f af[2], bf[5];
#pragma unroll
        for (int t = 0; t < 2; ++t) af[t] = *(const v2f*)(ak + 32 * t);
#pragma unroll
        for (int t = 0; t < 5; ++t) bf[t] = *(const v2f*)(bk + 32 * t);
#pragma unroll
        for (int i = 0; i < 2; ++i)
#pragma unroll
            for (int j = 0; j < 5; ++j)
                acc[i][j] = __builtin_amdgcn_wmma_f32_16x16x4_f32(
                    false, af[i], false, bf[j], (short)0, acc[i][j], false, false);
        ak += 4 * NN;   // k += 4  ->  two packed rows of 1600 floats
        bk += 4 * NN;
    }
    // C layout: VGPR r -> M = r (lanes 0-15), M = r+8 (lanes 16-31); N = l16
#pragma unroll
    for (int i = 0; i < 2; ++i)
#pragma unroll
        for (int j = 0; j < 5; ++j)
#pragma unroll
            for (int r = 0; r < 8; ++r)
                c[(size_t)(m0 + 16 * i + r + 8 * half) * NN + n0 + 16 * j + l16] = acc[i][j][r];
}

// ---------------- stage E: Y = [Xout, ind] @ w3 + b3 (in-place over Xout planes) ----------------
__global__ void y_kernel(float* __restrict__ XY, const float* __restrict__ ind,
                         const float* __restrict__ w3, const float* __restrict__ b3) {
    __shared__ float ws[33 * HC];
    for (int t = threadIdx.x; t < 33 * HC; t += blockDim.x) ws[t] = w3[t];
    __syncthreads();
    int p = blockIdx.x * blockDim.x + threadIdx.x;
    if (p >= PLANE) return;
    float xv[HC];
#pragma unroll
    for (int cc = 0; cc < HC; ++cc) xv[cc] = XY[(size_t)cc * PLANE + p];
    float indv = ind[p];
#pragma unroll 4
    for (int dd = 0; dd < HC; ++dd) {
        float acc = b3[dd] + indv * ws[HC * HC + dd];
#pragma unroll
        for (int cc = 0; cc < HC; ++cc) acc += xv[cc] * ws[cc * HC + dd];
        XY[(size_t)dd * PLANE + p] = acc;
    }
}

// ---------------- stage F: masked graph-norm statistics ----------------
__global__ void stat_sum(const float* __restrict__ Y, const unsigned char* __restrict__ U,
                         float* __restrict__ sums) {
    int d = blockIdx.x, slice = blockIdx.y;   // 32 x 8
    const float* y = Y + (size_t)d * PLANE;
    float s = 0.f;
    for (int p = slice * 80000 + threadIdx.x; p < (slice + 1) * 80000; p += 256)
        s += U[p] ? y[p] : 0.f;
    s = blockReduceSum256(s);
    if (threadIdx.x == 0) atomicAdd(&sums[d], s);
}

__global__ void stat_var(const float* __restrict__ Y, const unsigned char* __restrict__ U,
                         const float* __restrict__ sums, const float* __restrict__ cntp,
                         const float* __restrict__ ga, float* __restrict__ vars) {
    int d = blockIdx.x, slice = blockIdx.y;
    float cnt = fmaxf(cntp[0], 1.f);
    float am = ga[d] * sums[d] / cnt;
    const float* y = Y + (size_t)d * PLANE;
    float s = 0.f;
    for (int p = slice * 80000 + threadIdx.x; p < (slice + 1) * 80000; p += 256) {
        if (U[p]) { float xc = y[p] - am; s += xc * xc; }
    }
    s = blockReduceSum256(s);
    if (threadIdx.x == 0) atomicAdd(&vars[d], s);
}

// ---------------- stage G: gather + final dot ----------------
__global__ void out_kernel(const float* __restrict__ Y, const unsigned char* __restrict__ U,
                           const float* __restrict__ h,
                           const int* __restrict__ pos1, const int* __restrict__ pos2,
                           const float* __restrict__ sums, const float* __restrict__ vars,
                           const float* __restrict__ cntp,
                           const float* __restrict__ gw, const float* __restrict__ gb,
                           const float* __restrict__ ga,
                           const float* __restrict__ wd, const float* __restrict__ bd,
                           float* __restrict__ out) {
    int k = blockIdx.x * blockDim.x + threadIdx.x;
    if (k >= NP) return;
    int p = pos1[pos2[2 * k] * 2];         // pos1[pos2][:,0] pairs
    int q = pos1[pos2[2 * k + 1] * 2];
    float cnt = fmaxf(cntp[0], 1.f);
    int pq = p * NN + q, qp = q * NN + p;
    float mpq = U[pq] ? 1.f : 0.f;
    float mqp = U[qp] ? 1.f : 0.f;
    float acc = bd[0];
#pragma unroll 4
    for (int d = 0; d < HC; ++d) {
        float m = sums[d] / cnt;
        float rstd = rsqrtf(vars[d] / cnt + 1e-5f);
        const float* y = Y + (size_t)d * PLANE;
        float y1 = (gw[d] * (y[pq] - ga[d] * m) * rstd + gb[d]) * mpq;
        float y2 = (gw[d] * (y[qp] - ga[d] * m) * rstd + gb[d]) * mqp;
        acc += fmaxf(y1, 0.f) * fmaxf(y2, 0.f) * wd[d];
    }
#pragma unroll 4
    for (int d = 0; d < HC; ++d)
        acc += h[p * HC + d] * h[q * HC + d] * wd[HC + d];
    out[k] = acc;
}

// ---------------- host launch ----------------
extern "C" void kernel_launch(void* const* d_in, const int* in_sizes, int n_in,
                              void* d_out, int out_size, void* d_ws, size_t ws_size,
                              hipStream_t stream) {
    const int*   x     = (const int*)d_in[0];
    const int*   ei    = (const int*)d_in[1];
    const int*   pos1  = (const int*)d_in[2];
    const int*   pos2  = (const int*)d_in[3];
    const float* emb   = (const float*)d_in[4];
    const float* gcn_w = (const float*)d_in[5];
    const float* gcn_b = (const float*)d_in[6];
    const float* gn1_w = (const float*)d_in[7];
    const float* gn1_b = (const float*)d_in[8];
    const float* gn1_a = (const float*)d_in[9];
    const float* w1    = (const float*)d_in[10];
    const float* b1    = (const float*)d_in[11];
    const float* w2    = (const float*)d_in[12];
    const float* b2    = (const float*)d_in[13];
    const float* w3    = (const float*)d_in[14];
    const float* b3    = (const float*)d_in[15];
    const float* gn3_w = (const float*)d_in[16];
    const float* gn3_b = (const float*)d_in[17];
    const float* gn3_a = (const float*)d_in[18];
    const float* wd    = (const float*)d_in[19];
    const float* bd    = (const float*)d_in[20];

    // workspace carve-up (floats)
    float* ws = (float*)d_ws;
    float* At   = ws;                       // 32 * PLANE  (packed A)
    float* Mp   = At + (size_t)32 * PLANE;  // 32 * PLANE  (packed B)
    float* Xout = Mp + (size_t)32 * PLANE;  // 32 * PLANE  (becomes Y in-place)
    float* hv   = Xout + (size_t)32 * PLANE;    // EE*32
    float* mulb = hv + EE * HC;                 // EE*32
    float* ind  = mulb + EE * HC;               // PLANE
    unsigned int*  Abits = (unsigned int*)(ind + PLANE);       // 800*25 words
    unsigned char* U     = (unsigned char*)(Abits + 800 * 25); // PLANE bytes
    float* h    = (float*)(U + PLANE);
    float* hwb  = h + NN * HC;
    float* agg  = hwb + NN * HC;
    float* deg  = agg + NN * HC;
    float* sums = deg + NN;        // 32
    float* vars = sums + HC;       // 32
    float* cnt  = vars + HC;       // 1

    auto g = [](int n, int b) { return (n + b - 1) / b; };

    // init
    zero_f32<<<4096, 256, 0, stream>>>(At, (long)64 * PLANE);           // At + Mp
    zero_f32<<<g(PLANE, 256), 256, 0, stream>>>(ind, PLANE);
    zero_f32<<<g(800 * 25, 256), 256, 0, stream>>>((float*)Abits, 800 * 25);
    zero_f32<<<1, 256, 0, stream>>>(sums, 65);
    fill_f32<<<g(NN, 256), 256, 0, stream>>>(deg, NN, 1.0f);

    // GCN
    emb_kernel<<<g(NN * HC, 256), 256, 0, stream>>>(x, emb, h);
    deg_kernel<<<g(EE, 256), 256, 0, stream>>>(ei, deg);
    for (int l = 0; l < 2; ++l) {
        matmul32<<<g(NN * HC, 256), 256, 0, stream>>>(h, gcn_w + l * HC * HC, hwb);
        agg_init<<<g(NN * HC, 256), 256, 0, stream>>>(hwb, deg, gcn_b + l * HC, agg);
        scatter_gcn<<<g(EE, 256), 256, 0, stream>>>(ei, hwb, deg, agg);
        gn_relu_kernel<<<32, 256, 0, stream>>>(agg, gn1_w + l * HC, gn1_b + l * HC,
                                               gn1_a + l * HC, h);
    }

    // edge MLPs + dense scatter into packed GEMM layouts
    edge_mlp<<<g(EE * HC, 256), 256, 0, stream>>>(ei, h, w1, b1, w2, b2, hv, mulb);
    scatter_dense<<<g(EE * HC, 256), 256, 0, stream>>>(ei, hv, mulb, At, Mp, ind, Abits);

    // masks
    u_kernel<<<g(PLANE, 256), 256, 0, stream>>>(Abits, U);
    cnt_kernel<<<64, 256, 0, stream>>>(U, cnt);

    // the big one: 32 x (800x800x800) fp32 GEMM on the matrix pipe
    dim3 eg(25, 5, 32);
    einsum_wmma<<<eg, 64, 0, stream>>>(At, Mp, Xout);

    // Y = [Xout, ind] @ w3 + b3 (in-place), masked graph-norm stats, final gather
    y_kernel<<<g(PLANE, 256), 256, 0, stream>>>(Xout, ind, w3, b3);
    dim3 sg(32, 8);
    stat_sum<<<sg, 256, 0, stream>>>(Xout, U, sums);
    stat_var<<<sg, 256, 0, stream>>>(Xout, U, sums, cnt, gn3_a, vars);
    out_kernel<<<g(NP, 256), 256, 0, stream>>>(Xout, U, h, pos1, pos2, sums, vars, cnt,
                                               gn3_w, gn3_b, gn3_a, wd, bd, (float*)d_out);
}